// AttLayer_model_3_46755013984606
// MI455X (gfx1250) — compile-verified
//
#include <hip/hip_runtime.h>
#include <hip/hip_bf16.h>
#include <math.h>

typedef __attribute__((ext_vector_type(2))) float v2f;
typedef __attribute__((ext_vector_type(8))) float v8f;

#define B_  32
#define T_  2048
#define D_  1024
#define H_  5
#define BT_ (B_ * T_)
#define TSPLIT 16

// ---------------------------------------------------------------------------
// Kernel 1: scores[m] = sum_h uw[h] * tanh(x_fea[m]*W_fea[h] + (x_temp@W_temp)[m,h] + bw[h])
// WMMA f32 16x16x4: 16-row tiles, N=16 (H padded; cols 5..15 garbage, never read),
// K loop over D in steps of 4. One tile per wave, 4096 tiles total.
// ---------------------------------------------------------------------------
__global__ __launch_bounds__(256) void score_kernel(
    const float* __restrict__ x_temp, const float* __restrict__ x_fea,
    const float* __restrict__ W_temp, const float* __restrict__ W_fea,
    const float* __restrict__ bw,     const float* __restrict__ uw,
    float* __restrict__ scores)
{
    __shared__ float Wt[H_ * D_];   // transposed W_temp: Wt[n*D + k] = W_temp[k*H + n]
    const int tid = threadIdx.x;
    for (int e = tid; e < H_ * D_; e += 256) {
        int n = e / D_, k = e % D_;
        Wt[e] = W_temp[k * H_ + n];
    }
    __syncthreads();

    const int lane  = tid & 31;
    const int wave  = tid >> 5;
    const int tile  = blockIdx.x * 8 + wave;       // 512 blocks * 8 waves = 4096 tiles
    const int m0    = tile * 16;
    const int colm  = lane & 15;                   // M for A-frag, N/col for B/C-frag
    const int khalf = (lane >> 4) * 2;             // upper lane half holds K+2,K+3

    const float* arow = x_temp + (size_t)(m0 + colm) * D_;
    const float* brow = Wt + (colm < H_ ? colm : (H_ - 1)) * D_;  // clamp: junk cols ignored

    v8f c = {0.f, 0.f, 0.f, 0.f, 0.f, 0.f, 0.f, 0.f};
    #pragma unroll 4
    for (int kb = 0; kb < D_; kb += 4) {
        v2f a = *(const v2f*)(arow + kb + khalf);  // A[m, kb+khalf .. +1]
        v2f b = *(const v2f*)(brow + kb + khalf);  // B[kb+khalf .. +1, n]
        c = __builtin_amdgcn_wmma_f32_16x16x4_f32(
                /*neg_a=*/false, a, /*neg_b=*/false, b,
                /*c_mod=*/(short)0, c, /*reuse_a=*/false, /*reuse_b=*/false);
    }

    // Epilogue: C/D layout -> VGPR i holds row (m0 + i + 8*(lane/16)), col = lane&15
    const float wf  = (colm < H_) ? W_fea[colm] : 0.0f;
    const float bwc = (colm < H_) ? bw[colm]    : 0.0f;
    const float uwc = (colm < H_) ? uw[colm]    : 0.0f;
    #pragma unroll
    for (int i = 0; i < 8; ++i) {
        const int row = m0 + i + 8 * (lane >> 4);
        float v = (colm < H_) ? uwc * tanhf(x_fea[row] * wf + c[i] + bwc) : 0.0f;
        v += __shfl_xor(v, 1, 16);
        v += __shfl_xor(v, 2, 16);
        v += __shfl_xor(v, 4, 16);
        v += __shfl_xor(v, 8, 16);
        if (colm == 0) scores[row] = v;
    }
}

// ---------------------------------------------------------------------------
// Kernel 2: fused masked softmax per batch:
//   w[t] = exp(s[t]-max)*mask[t] / sum_t exp(s[t]-max)*mask[t]
// (the plain-softmax denominator cancels against the mask renorm)
// ---------------------------------------------------------------------------
__global__ __launch_bounds__(256) void softmax_kernel(
    const float* __restrict__ scores, const int* __restrict__ mask,
    float* __restrict__ weights)
{
    __shared__ float red[256];
    const int b   = blockIdx.x;
    const int tid = threadIdx.x;
    const float* s = scores + (size_t)b * T_;
    const int*   m = mask   + (size_t)b * T_;

    float local[8];
    float mx = -3.4e38f;
    #pragma unroll
    for (int i = 0; i < 8; ++i) { local[i] = s[tid + i * 256]; mx = fmaxf(mx, local[i]); }
    red[tid] = mx; __syncthreads();
    for (int st = 128; st > 0; st >>= 1) {
        if (tid < st) red[tid] = fmaxf(red[tid], red[tid + st]);
        __syncthreads();
    }
    mx = red[0]; __syncthreads();

    float e[8]; float sum = 0.f;
    #pragma unroll
    for (int i = 0; i < 8; ++i) {
        e[i] = expf(local[i] - mx) * (m[tid + i * 256] ? 1.0f : 0.0f);
        sum += e[i];
    }
    red[tid] = sum; __syncthreads();
    for (int st = 128; st > 0; st >>= 1) {
        if (tid < st) red[tid] += red[tid + st];
        __syncthreads();
    }
    const float inv = 1.0f / red[0];
    #pragma unroll
    for (int i = 0; i < 8; ++i)
        weights[(size_t)b * T_ + tid + i * 256] = e[i] * inv;
}

// ---------------------------------------------------------------------------
// Kernel 3a: partial[ts,b,d] = sum_{t in slice ts} w[b,t]*x_temp[b,t,d]
// One block per (b, ts); each thread owns 4 d's -> one float4 load covers D=1024.
// Deterministic split-T (no float atomics).
// ---------------------------------------------------------------------------
__global__ __launch_bounds__(256) void pool_partial_kernel(
    const float* __restrict__ x_temp, const float* __restrict__ weights,
    float* __restrict__ partial)
{
    const int b  = blockIdx.x;
    const int ts = blockIdx.y;
    const int d  = threadIdx.x * 4;
    const int chunk = T_ / TSPLIT;
    const int t0 = ts * chunk;

    const float* w  = weights + (size_t)b * T_;
    const float* xb = x_temp + (size_t)b * T_ * D_ + d;

    float4 acc = make_float4(0.f, 0.f, 0.f, 0.f);
    for (int t = t0; t < t0 + chunk; ++t) {
        int tp = t + 4; if (tp > T_ - 1) tp = T_ - 1;
        __builtin_prefetch(xb + (size_t)tp * D_, 0, 0);   // global_prefetch_b8
        const float  wt = w[t];
        const float4 x  = *(const float4*)(xb + (size_t)t * D_);
        acc.x += wt * x.x; acc.y += wt * x.y; acc.z += wt * x.z; acc.w += wt * x.w;
    }
    *(float4*)(partial + (size_t)(ts * B_ + b) * D_ + d) = acc;
}

// ---------------------------------------------------------------------------
// Kernel 3b: out[b,d] = sum_ts partial[ts,b,d]   (writes every output element)
// ---------------------------------------------------------------------------
__global__ __launch_bounds__(256) void pool_reduce_kernel(
    const float* __restrict__ partial, float* __restrict__ out)
{
    const int idx = blockIdx.x * 256 + threadIdx.x;  // over B*D/4
    const int bd  = idx * 4;
    const int b   = bd / D_;
    const int d   = bd % D_;
    float4 acc = make_float4(0.f, 0.f, 0.f, 0.f);
    #pragma unroll
    for (int ts = 0; ts < TSPLIT; ++ts) {
        const float4 p = *(const float4*)(partial + (size_t)(ts * B_ + b) * D_ + d);
        acc.x += p.x; acc.y += p.y; acc.z += p.z; acc.w += p.w;
    }
    *(float4*)(out + bd) = acc;
}

// ---------------------------------------------------------------------------
extern "C" void kernel_launch(void* const* d_in, const int* in_sizes, int n_in,
                              void* d_out, int out_size, void* d_ws, size_t ws_size,
                              hipStream_t stream) {
    const float* x_temp = (const float*)d_in[0];
    const float* x_fea  = (const float*)d_in[1];
    const int*   mask   = (const int*)  d_in[2];
    const float* W_temp = (const float*)d_in[3];
    const float* W_fea  = (const float*)d_in[4];
    const float* bw     = (const float*)d_in[5];
    const float* uw     = (const float*)d_in[6];
    float* out = (float*)d_out;

    float* ws      = (float*)d_ws;
    float* scores  = ws;                 // BT floats
    float* weights = ws + BT_;           // BT floats
    float* partial = ws + 2 * BT_;       // TSPLIT*B*D floats

    score_kernel<<<BT_ / 16 / 8, 256, 0, stream>>>(x_temp, x_fea, W_temp, W_fea, bw, uw, scores);
    softmax_kernel<<<B_, 256, 0, stream>>>(scores, mask, weights);
    pool_partial_kernel<<<dim3(B_, TSPLIT), 256, 0, stream>>>(x_temp, weights, partial);
    pool_reduce_kernel<<<(B_ * D_ / 4) / 256, 256, 0, stream>>>(partial, out);
}